// NTXentLoss_21852793602889
// MI455X (gfx1250) — compile-verified
//
#include <hip/hip_runtime.h>
#include <math.h>
#include <stdint.h>

typedef __attribute__((ext_vector_type(2))) float v2f;
typedef __attribute__((ext_vector_type(8))) float v8f;

#define NROWS 8192   // 2B
#define DDIM  256
#define HALF_B 4096
#define MTILE 128    // macro-tile edge
#define KC    32     // K staging chunk
#define LDST  36     // padded LDS row stride (32 + 4) -> bank-conflict free
#define CHUNKF (MTILE * LDST)          // floats per buffer (4608)
#define CHUNKB (CHUNKF * 4)            // bytes per buffer (18432)

// ---------------------------------------------------------------------------
// CDNA5 async global->LDS copy (16B per lane), tracked by ASYNCcnt.
// ---------------------------------------------------------------------------
__device__ __forceinline__ void async_ld16(uint32_t lds_off, const float* gaddr) {
    asm volatile("global_load_async_to_lds_b128 %0, %1, off"
                 :: "v"(lds_off), "v"(gaddr)
                 : "memory");
}
__device__ __forceinline__ void wait_async0() {
    asm volatile("s_wait_asynccnt 0x0" ::: "memory");
}

// ---------------------------------------------------------------------------
// Kernel 1: row-normalize concat(z1,z2) -> zn; zero rowsum/pos accumulators.
// ---------------------------------------------------------------------------
__global__ __launch_bounds__(256) void ntx_normalize(const float* __restrict__ z1,
                                                     const float* __restrict__ z2,
                                                     float* __restrict__ zn,
                                                     float* __restrict__ rowsum,
                                                     float* __restrict__ pos) {
    const int row = blockIdx.x;
    const int t   = threadIdx.x;
    const float* src = (row < HALF_B) ? (z1 + (size_t)row * DDIM)
                                      : (z2 + (size_t)(row - HALF_B) * DDIM);
    float v = src[t];

    __shared__ float red[256];
    red[t] = v * v;
    __syncthreads();
    #pragma unroll
    for (int s = 128; s > 0; s >>= 1) {
        if (t < s) red[t] += red[t + s];
        __syncthreads();
    }
    float nrm = fmaxf(sqrtf(red[0]), 1e-8f);
    zn[(size_t)row * DDIM + t] = v / nrm;

    if (t == 0) {
        rowsum[row] = 0.0f;
        pos[row]    = 0.0f;
    }
}

// ---------------------------------------------------------------------------
// Kernel 2: tiled zn @ zn^T with V_WMMA_F32_16X16X4_F32; streamed logsumexp.
// 256 threads (8 waves) per 128x128 tile; double-buffered async LDS staging.
// ---------------------------------------------------------------------------
__global__ __launch_bounds__(256) void ntx_sim_lse(const float* __restrict__ zn,
                                                   float* __restrict__ rowsum,
                                                   float* __restrict__ pos) {
    __shared__ float As[2 * CHUNKF];   // 36 KB
    __shared__ float Bs[2 * CHUNKF];   // 36 KB

    const int i0   = blockIdx.y * MTILE;
    const int j0   = blockIdx.x * MTILE;
    const int tid  = threadIdx.x;
    const int lane = tid & 31;
    const int w    = tid >> 5;       // wave id 0..7
    const int ln   = lane & 15;      // N / M-low index within fragment
    const int hb   = lane >> 4;      // half-wave select
    const int koff = hb << 1;        // K offset {0,2} per ISA A/B layout

    // ---- per-thread staging addresses: 4 float4 slots each for A and B ----
    const float* gA[4];
    const float* gB[4];
    uint32_t     lA[4];
    uint32_t     lB[4];
    const uint32_t baseA = (uint32_t)(uintptr_t)(void*)As;
    const uint32_t baseB = (uint32_t)(uintptr_t)(void*)Bs;
    #pragma unroll
    for (int it = 0; it < 4; ++it) {
        int f = tid + it * 256;        // float4 slot 0..1023
        int r = f >> 3;                // row 0..127
        int c = (f & 7) << 2;          // col 0,4,...,28
        gA[it] = zn + (size_t)(i0 + r) * DDIM + c;
        gB[it] = zn + (size_t)(j0 + r) * DDIM + c;
        lA[it] = baseA + (uint32_t)(r * LDST + c) * 4u;
        lB[it] = baseB + (uint32_t)(r * LDST + c) * 4u;
    }

    v8f acc[8];
    v8f zero = {};
    #pragma unroll
    for (int t = 0; t < 8; ++t) acc[t] = zero;

    // prologue: stage chunk 0 into buffer 0
    #pragma unroll
    for (int it = 0; it < 4; ++it) {
        async_ld16(lA[it], gA[it]);
        async_ld16(lB[it], gB[it]);
    }

    const int NCH = DDIM / KC;   // 8 chunks
    for (int c = 0; c < NCH; ++c) {
        const int buf = c & 1;
        wait_async0();           // this wave's loads into buf have landed
        __syncthreads();         // all waves landed; prior reads of buf^1 done

        // prefetch next chunk into the other buffer (overlaps WMMA below)
        if (c + 1 < NCH) {
            const uint32_t bo = (uint32_t)((buf ^ 1) * CHUNKB);
            const int kn = (c + 1) * KC;
            #pragma unroll
            for (int it = 0; it < 4; ++it) {
                async_ld16(lA[it] + bo, gA[it] + kn);
                async_ld16(lB[it] + bo, gB[it] + kn);
            }
        }

        // ---- 8 k-steps of 4 over this chunk ----
        const float* A  = As + buf * CHUNKF;
        const float* Bb = Bs + buf * CHUNKF;
        const float* aPtr = A + (16 * w + ln) * LDST + koff;
        #pragma unroll
        for (int kk = 0; kk < KC; kk += 4) {
            v2f a = *(const v2f*)(aPtr + kk);
            v2f bf[8];
            #pragma unroll
            for (int t = 0; t < 8; ++t)
                bf[t] = *(const v2f*)(Bb + (16 * t + ln) * LDST + kk + koff);
            #pragma unroll
            for (int t = 0; t < 8; ++t)
                acc[t] = __builtin_amdgcn_wmma_f32_16x16x4_f32(
                    false, a, false, bf[t], (short)0, acc[t], false, false);
        }
        __syncthreads();
    }

    // ---- epilogue: scale by 1/T = 2, diag-mask, positive gather, exp-sum ----
    float s[8];
    #pragma unroll
    for (int r = 0; r < 8; ++r) s[r] = 0.0f;

    #pragma unroll
    for (int t = 0; t < 8; ++t) {
        const int gj = j0 + 16 * t + ln;
        #pragma unroll
        for (int r = 0; r < 8; ++r) {
            const int gi = i0 + 16 * w + r + 8 * hb;
            float v = acc[t][r] * 2.0f;                 // sim / TEMPERATURE
            if (gj == (gi ^ HALF_B)) pos[gi] = v;       // unique writer per row
            float e = (gi == gj) ? 0.0f : __expf(v);    // diag -> exp(NEG_FILL)=0
            s[r] += e;
        }
    }

    // reduce across the 16 lanes of each half-wave (same row), then atomic add
    #pragma unroll
    for (int r = 0; r < 8; ++r) {
        float x = s[r];
        x += __shfl_xor(x, 1, 32);
        x += __shfl_xor(x, 2, 32);
        x += __shfl_xor(x, 4, 32);
        x += __shfl_xor(x, 8, 32);
        if (ln == 0) {
            atomicAdd(&rowsum[i0 + 16 * w + r + 8 * hb], x);
        }
    }
}

// ---------------------------------------------------------------------------
// Kernel 3: loss = mean_i( log(rowsum[i]) - pos[i] )
// ---------------------------------------------------------------------------
__global__ __launch_bounds__(256) void ntx_loss(const float* __restrict__ rowsum,
                                                const float* __restrict__ pos,
                                                float* __restrict__ out) {
    __shared__ float red[256];
    const int t = threadIdx.x;
    float a = 0.0f;
    for (int i = t; i < NROWS; i += 256) {
        a += __logf(rowsum[i]) - pos[i];
    }
    red[t] = a;
    __syncthreads();
    #pragma unroll
    for (int s = 128; s > 0; s >>= 1) {
        if (t < s) red[t] += red[t + s];
        __syncthreads();
    }
    if (t == 0) out[0] = red[0] / (float)NROWS;
}

extern "C" void kernel_launch(void* const* d_in, const int* in_sizes, int n_in,
                              void* d_out, int out_size, void* d_ws, size_t ws_size,
                              hipStream_t stream) {
    const float* z1 = (const float*)d_in[0];
    const float* z2 = (const float*)d_in[1];

    float* zn     = (float*)d_ws;                      // 8192*256 floats (8 MB)
    float* rowsum = zn + (size_t)NROWS * DDIM;         // 8192 floats
    float* pos    = rowsum + NROWS;                    // 8192 floats

    ntx_normalize<<<NROWS, 256, 0, stream>>>(z1, z2, zn, rowsum, pos);

    dim3 grid(NROWS / MTILE, NROWS / MTILE);           // 64 x 64 macro-tiles
    ntx_sim_lse<<<grid, 256, 0, stream>>>(zn, rowsum, pos);

    ntx_loss<<<1, 256, 0, stream>>>(rowsum, pos, (float*)d_out);
}